// MPNN_54631984005154
// MI455X (gfx1250) — compile-verified
//
#include <hip/hip_runtime.h>
#include <hip/hip_bf16.h>

#define N_NODES  50000
#define N_EDGES  600000
#define N_GRAPHS 512
#define IN_NODE  32
#define IN_EDGE  16
#define EMB      128
#define HID      64
#define N_CLS    10
#define N_LAYERS 5
#define BN_EPS   1e-5f

typedef __attribute__((ext_vector_type(2))) float v2f;
typedef __attribute__((ext_vector_type(8))) float v8f;

#define LDS_KP 132   // K stride pad (+4 floats) -> 16 lanes hit distinct LDS banks

// ---------------------------------------------------------------------------
// fp32 WMMA GEMM: C[M,N] = A[M,K] @ B[K,N] + bias[N]
// Requires M%16==0, N%16==0, K%4==0, K<=128.
// block = 128 threads (4 waves); grid = (M/16, ceil(N/64)).
// The 16xK A-tile is shared by all 4 waves of the block -> staged once in LDS
// (pad stride 132 so each half-wave's 16 ds_load_b64 reads are conflict-free).
// Each wave then computes one 16x16 tile with V_WMMA_F32_16X16X4_F32.
// A layout (16x4 f32): lane&15 = M row, lane>>4 selects K pair {0,1} vs {2,3}.
// B layout (4x16 f32): lane&15 = N col, lane>>4 selects K row pair.
// C layout (16x16 f32): VGPR r holds row r + 8*(lane>>4), col = lane&15.
// ---------------------------------------------------------------------------
__global__ void gemm_bias_wmma(const float* __restrict__ A,
                               const float* __restrict__ B,
                               const float* __restrict__ bias,
                               float* __restrict__ C,
                               int M, int N, int K) {
  __shared__ float As[16 * LDS_KP];

  const int tid  = threadIdx.x;
  const int wave = tid >> 5;
  const int lane = tid & 31;
  const int half = lane >> 4;      // 0 or 1
  const int l16  = lane & 15;
  const int m0   = blockIdx.x << 4;
  const int n0   = (blockIdx.y << 6) + (wave << 4);

  // cooperative fill of the 16xK A tile (coalesced along K)
  const int tot = K << 4;          // 16 * K
  for (int idx = tid; idx < tot; idx += 128) {
    const int r  = idx / K;
    const int kk = idx - r * K;
    As[r * LDS_KP + kk] = A[(long long)(m0 + r) * K + kk];
  }
  __syncthreads();

  if (n0 < N) {                    // wave-uniform; EXEC all-ones inside
    const float* Alds = As + l16 * LDS_KP;
    v8f acc = {};
    for (int k = 0; k < K; k += 4) {
      const int ka = k + (half << 1);
      v2f a, b;
      a.x = Alds[ka];
      a.y = Alds[ka + 1];
      b.x = B[(long long)ka       * N + n0 + l16];
      b.y = B[(long long)(ka + 1) * N + n0 + l16];
      // 8 args: (neg_a, A, neg_b, B, c_mod, C, reuse_a, reuse_b)
      acc = __builtin_amdgcn_wmma_f32_16x16x4_f32(
          false, a, false, b, (short)0, acc, false, false);
    }
    const float bv = bias[n0 + l16];
#pragma unroll
    for (int r = 0; r < 8; ++r) {
      const int m = m0 + r + (half << 3);
      C[(long long)m * N + n0 + l16] = acc[r] + bv;
    }
  }
}

// ---------------------------------------------------------------------------
// Degree: deg[row[e]] += 1 (atomic), then deg += 1.0 and dinv = rsqrt(deg)
// ---------------------------------------------------------------------------
__global__ void deg_count(const long long* __restrict__ row,
                          float* __restrict__ deg, int nE) {
  int i = blockIdx.x * blockDim.x + threadIdx.x;
  if (i < nE) atomicAdd(&deg[row[i]], 1.0f);
}

__global__ void deg_finalize(float* __restrict__ deg,
                             float* __restrict__ dinv, int n) {
  int i = blockIdx.x * blockDim.x + threadIdx.x;
  if (i < n) {
    float d = deg[i] + 1.0f;   // reference: segment_sum + 1.0
    deg[i]  = d;
    dinv[i] = rsqrtf(d);
  }
}

// ---------------------------------------------------------------------------
// Edge messages: one wave per edge, lane handles 4 channels (float4).
// agg[col] += dinv[row]*dinv[col] * relu(hW[row] + e[edge])   (f32 atomics)
// Prefetch a lookahead edge's hW row (global_prefetch_b8) to warm L2 ahead
// of the random gather that bounds this kernel.
// ---------------------------------------------------------------------------
#define EDGE_LOOKAHEAD 64

__global__ void edge_msg(const float* __restrict__ hW,
                         const float* __restrict__ e,
                         const long long* __restrict__ row,
                         const long long* __restrict__ col,
                         const float* __restrict__ dinv,
                         float* __restrict__ agg, int nE) {
  const int eidx = blockIdx.x * (blockDim.x >> 5) + (threadIdx.x >> 5);
  if (eidx >= nE) return;
  const int lane = threadIdx.x & 31;

  const int epf = eidx + EDGE_LOOKAHEAD;
  if (epf < nE) {
    const long long rpf = row[epf];
    __builtin_prefetch(hW + rpf * EMB + lane * 4, 0, 1);
  }

  const long long r = row[eidx];
  const long long c = col[eidx];
  const float norm = dinv[r] * dinv[c];

  const float4 h4 = ((const float4*)(hW + r * EMB))[lane];
  const float4 e4 = ((const float4*)(e + (long long)eidx * EMB))[lane];
  float4 m;
  m.x = fmaxf(h4.x + e4.x, 0.0f) * norm;
  m.y = fmaxf(h4.y + e4.y, 0.0f) * norm;
  m.z = fmaxf(h4.z + e4.z, 0.0f) * norm;
  m.w = fmaxf(h4.w + e4.w, 0.0f) * norm;

  float* dst = agg + c * EMB + lane * 4;
  atomicAdd(dst + 0, m.x);
  atomicAdd(dst + 1, m.y);
  atomicAdd(dst + 2, m.z);
  atomicAdd(dst + 3, m.w);
}

// ---------------------------------------------------------------------------
// Self term: agg[i] += relu(hW[i] + root) / deg[i]   (one wave per node)
// ---------------------------------------------------------------------------
__global__ void self_term(const float* __restrict__ hW,
                          const float* __restrict__ root,
                          const float* __restrict__ deg,
                          float* __restrict__ agg, int nN) {
  const int node = blockIdx.x * (blockDim.x >> 5) + (threadIdx.x >> 5);
  if (node >= nN) return;
  const int lane = threadIdx.x & 31;
  const float invd = 1.0f / deg[node];
  const float4 h4 = ((const float4*)(hW + (long long)node * EMB))[lane];
  const float4 r4 = ((const float4*)root)[lane];
  float4* av = (float4*)(agg + (long long)node * EMB);
  float4 a4 = av[lane];
  a4.x += fmaxf(h4.x + r4.x, 0.0f) * invd;
  a4.y += fmaxf(h4.y + r4.y, 0.0f) * invd;
  a4.z += fmaxf(h4.z + r4.z, 0.0f) * invd;
  a4.w += fmaxf(h4.w + r4.w, 0.0f) * invd;
  av[lane] = a4;
}

// ---------------------------------------------------------------------------
// BN statistics: per-channel sum & sumsq (blockDim.x == cdim; rows striped
// across blocks; coalesced full-row loads; atomic accumulate into [cdim]).
// ---------------------------------------------------------------------------
__global__ void bn_stats(const float* __restrict__ x,
                         float* __restrict__ sum,
                         float* __restrict__ sumsq,
                         int n, int cdim) {
  const int c = threadIdx.x;
  float s = 0.0f, s2 = 0.0f;
  for (int i = blockIdx.x; i < n; i += gridDim.x) {
    float v = x[(long long)i * cdim + c];
    s  += v;
    s2 += v * v;
  }
  atomicAdd(&sum[c], s);
  atomicAdd(&sumsq[c], s2);
}

// ---------------------------------------------------------------------------
// BN apply (+ optional ReLU, optional residual add into dst)
// ---------------------------------------------------------------------------
__global__ void bn_apply(const float* __restrict__ x,
                         const float* __restrict__ sum,
                         const float* __restrict__ sumsq,
                         const float* __restrict__ g,
                         const float* __restrict__ b,
                         float* dst, int n, int cdim,
                         int do_relu, int do_res) {
  const long long gid = (long long)blockIdx.x * blockDim.x + threadIdx.x;
  if (gid >= (long long)n * cdim) return;
  const int c = (int)(gid % cdim);
  const float inv_n = 1.0f / (float)n;
  const float m   = sum[c] * inv_n;
  const float var = fmaxf(sumsq[c] * inv_n - m * m, 0.0f);
  float hn = (x[gid] - m) * rsqrtf(var + BN_EPS) * g[c] + b[c];
  if (do_relu) hn = fmaxf(hn, 0.0f);
  dst[gid] = do_res ? (hn + dst[gid]) : hn;
}

// ---------------------------------------------------------------------------
// Global mean pool: wave per node, float4 atomics into hg[graph], count graph.
// ---------------------------------------------------------------------------
__global__ void pool_sum(const float* __restrict__ h,
                         const long long* __restrict__ batch,
                         float* __restrict__ hg,
                         float* __restrict__ counts, int nN) {
  const int node = blockIdx.x * (blockDim.x >> 5) + (threadIdx.x >> 5);
  if (node >= nN) return;
  const int lane = threadIdx.x & 31;
  const long long gidx = batch[node];
  if (lane == 0) atomicAdd(&counts[gidx], 1.0f);
  const float4 v = ((const float4*)(h + (long long)node * EMB))[lane];
  float* dst = hg + gidx * EMB + lane * 4;
  atomicAdd(dst + 0, v.x);
  atomicAdd(dst + 1, v.y);
  atomicAdd(dst + 2, v.z);
  atomicAdd(dst + 3, v.w);
}

__global__ void pool_div(float* __restrict__ hg,
                         const float* __restrict__ counts) {
  const int gid = blockIdx.x * blockDim.x + threadIdx.x;  // N_GRAPHS*EMB
  const int g = gid / EMB;
  hg[gid] /= fmaxf(counts[g], 1.0f);
}

// ---------------------------------------------------------------------------
// Tiny scalar GEMM for the 64 -> 10 head (N not a multiple of 16)
// ---------------------------------------------------------------------------
__global__ void small_gemm_bias(const float* __restrict__ A,
                                const float* __restrict__ B,
                                const float* __restrict__ bias,
                                float* __restrict__ C,
                                int M, int N, int K) {
  const int gid = blockIdx.x * blockDim.x + threadIdx.x;
  if (gid >= M * N) return;
  const int m = gid / N, n = gid % N;
  float s = bias[n];
  for (int k = 0; k < K; ++k) s += A[m * K + k] * B[k * N + n];
  C[gid] = s;
}

// ---------------------------------------------------------------------------
extern "C" void kernel_launch(void* const* d_in, const int* in_sizes, int n_in,
                              void* d_out, int out_size, void* d_ws, size_t ws_size,
                              hipStream_t stream) {
  const float* x         = (const float*)d_in[0];
  const long long* eidx  = (const long long*)d_in[1];
  const float* edge_attr = (const float*)d_in[2];
  const long long* batch = (const long long*)d_in[3];
  const float* node_W    = (const float*)d_in[4];
  const float* node_b    = (const float*)d_in[5];
  const float* edge_W    = (const float*)d_in[6];
  const float* edge_b    = (const float*)d_in[7];
  const float* lin_W     = (const float*)d_in[8];
  const float* lin_b     = (const float*)d_in[9];
  const float* root      = (const float*)d_in[10];
  const float* bn_g      = (const float*)d_in[11];
  const float* bn_b      = (const float*)d_in[12];
  const float* mlp_W0    = (const float*)d_in[13];
  const float* mlp_b0    = (const float*)d_in[14];
  const float* mlp_g0    = (const float*)d_in[15];
  const float* mlp_bb0   = (const float*)d_in[16];
  const float* mlp_W1    = (const float*)d_in[17];
  const float* mlp_b1    = (const float*)d_in[18];
  const float* mlp_g1    = (const float*)d_in[19];
  const float* mlp_bb1   = (const float*)d_in[20];

  const long long* row = eidx;            // source (j)
  const long long* col = eidx + N_EDGES;  // target (i)

  // workspace layout (floats)
  float* ws = (float*)d_ws;
  size_t off = 0;
  float* e_emb  = ws + off; off += (size_t)N_EDGES * EMB;   // 76.8 M
  float* h      = ws + off; off += (size_t)N_NODES * EMB;   // 6.4 M
  float* hW     = ws + off; off += (size_t)N_NODES * EMB;   // 6.4 M
  float* agg    = ws + off; off += (size_t)N_NODES * EMB;   // 6.4 M
  float* deg    = ws + off; off += N_NODES;
  float* dinv   = ws + off; off += N_NODES;
  float* bnsum  = ws + off; off += EMB;                     // reused for HID/N_CLS
  float* bnsq   = ws + off; off += EMB;                     // adjacent to bnsum
  float* hg     = ws + off; off += (size_t)N_GRAPHS * EMB;
  float* counts = ws + off; off += N_GRAPHS;                // adjacent to hg
  float* z0     = ws + off; off += (size_t)N_GRAPHS * HID;
  float* z1     = ws + off; off += (size_t)N_GRAPHS * N_CLS;
  (void)ws_size; (void)in_sizes; (void)n_in; (void)out_size;

  // ---- degrees (identical across layers: compute once) ----
  hipMemsetAsync(deg, 0, N_NODES * sizeof(float), stream);
  deg_count<<<(N_EDGES + 255) / 256, 256, 0, stream>>>(row, deg, N_EDGES);
  deg_finalize<<<(N_NODES + 255) / 256, 256, 0, stream>>>(deg, dinv, N_NODES);

  // ---- encoders (WMMA fp32 GEMMs) ----
  gemm_bias_wmma<<<dim3(N_EDGES / 16, EMB / 64), 128, 0, stream>>>(
      edge_attr, edge_W, edge_b, e_emb, N_EDGES, EMB, IN_EDGE);
  gemm_bias_wmma<<<dim3(N_NODES / 16, EMB / 64), 128, 0, stream>>>(
      x, node_W, node_b, h, N_NODES, EMB, IN_NODE);

  // ---- 5 GCN layers ----
  for (int l = 0; l < N_LAYERS; ++l) {
    gemm_bias_wmma<<<dim3(N_NODES / 16, EMB / 64), 128, 0, stream>>>(
        h, lin_W + (size_t)l * EMB * EMB, lin_b + (size_t)l * EMB,
        hW, N_NODES, EMB, EMB);

    hipMemsetAsync(agg, 0, (size_t)N_NODES * EMB * sizeof(float), stream);
    edge_msg<<<(N_EDGES + 7) / 8, 256, 0, stream>>>(
        hW, e_emb, row, col, dinv, agg, N_EDGES);
    self_term<<<(N_NODES + 7) / 8, 256, 0, stream>>>(
        hW, root + (size_t)l * EMB, deg, agg, N_NODES);

    hipMemsetAsync(bnsum, 0, 2 * EMB * sizeof(float), stream);
    bn_stats<<<512, EMB, 0, stream>>>(agg, bnsum, bnsq, N_NODES, EMB);
    bn_apply<<<((size_t)N_NODES * EMB + 255) / 256, 256, 0, stream>>>(
        agg, bnsum, bnsq, bn_g + (size_t)l * EMB, bn_b + (size_t)l * EMB,
        h, N_NODES, EMB, /*relu=*/(l != N_LAYERS - 1), /*res=*/1);
  }

  // ---- global mean pool ----
  hipMemsetAsync(hg, 0, ((size_t)N_GRAPHS * EMB + N_GRAPHS) * sizeof(float), stream);
  pool_sum<<<(N_NODES + 7) / 8, 256, 0, stream>>>(h, batch, hg, counts, N_NODES);
  pool_div<<<(N_GRAPHS * EMB) / 256, 256, 0, stream>>>(hg, counts);

  // ---- MLP head ----
  gemm_bias_wmma<<<dim3(N_GRAPHS / 16, 1), 128, 0, stream>>>(
      hg, mlp_W0, mlp_b0, z0, N_GRAPHS, HID, EMB);
  hipMemsetAsync(bnsum, 0, 2 * EMB * sizeof(float), stream);
  bn_stats<<<64, HID, 0, stream>>>(z0, bnsum, bnsq, N_GRAPHS, HID);
  bn_apply<<<((size_t)N_GRAPHS * HID + 255) / 256, 256, 0, stream>>>(
      z0, bnsum, bnsq, mlp_g0, mlp_bb0, z0, N_GRAPHS, HID, /*relu=*/1, /*res=*/0);

  small_gemm_bias<<<(N_GRAPHS * N_CLS + 255) / 256, 256, 0, stream>>>(
      z0, mlp_W1, mlp_b1, z1, N_GRAPHS, N_CLS, HID);
  hipMemsetAsync(bnsum, 0, 2 * EMB * sizeof(float), stream);
  bn_stats<<<64, N_CLS, 0, stream>>>(z1, bnsum, bnsq, N_GRAPHS, N_CLS);
  bn_apply<<<((size_t)N_GRAPHS * N_CLS + 255) / 256, 256, 0, stream>>>(
      z1, bnsum, bnsq, mlp_g1, mlp_bb1, (float*)d_out,
      N_GRAPHS, N_CLS, /*relu=*/0, /*res=*/0);
}